// GFNOBlock_6107443494944
// MI455X (gfx1250) — compile-verified
//
#include <hip/hip_runtime.h>
#include <hip/hip_bf16.h>
#include <math.h>

typedef float v2f __attribute__((ext_vector_type(2)));
typedef float v8f __attribute__((ext_vector_type(8)));

#define GN   131072   // nodes
#define GC   128      // channels
#define GB   64       // batches
#define GNPG 2048     // nodes per group
#define GM   64       // modes

static __device__ __forceinline__ v8f wmma4(v2f a, v2f b, v8f c) {
    // V_WMMA_F32_16X16X4_F32 : D(16x16 f32) = A(16x4) * B(4x16) + C
    return __builtin_amdgcn_wmma_f32_16x16x4_f32(
        /*neg_a=*/false, a, /*neg_b=*/false, b,
        /*c_mod=*/(short)0, c, /*reuse_a=*/false, /*reuse_b=*/false);
}

// ---------------- workspace zero ----------------
__global__ __launch_bounds__(256) void k_zero(float* __restrict__ p, long n) {
    long i = (long)blockIdx.x * blockDim.x + threadIdx.x;
    long stride = (long)gridDim.x * blockDim.x;
    for (; i < n; i += stride) p[i] = 0.0f;
}

// ---------------- edge aggregation: agg[dst] += h[src], deg[dst] += 1 ----------------
__global__ __launch_bounds__(256) void k_agg(const float* __restrict__ h,
                                             const int* __restrict__ edge,
                                             float* __restrict__ agg,
                                             float* __restrict__ deg, int E) {
    long gid = (long)blockIdx.x * 256 + threadIdx.x;
    int e = (int)(gid >> 5);
    if (e >= E) return;
    int c4 = ((int)gid & 31) * 4;
    int src = edge[e];
    int dst = edge[E + e];
    const float4 v = *(const float4*)(h + (size_t)src * GC + c4);
    float* base = agg + (size_t)dst * GC + c4;
    __hip_atomic_fetch_add(base + 0, v.x, __ATOMIC_RELAXED, __HIP_MEMORY_SCOPE_AGENT);
    __hip_atomic_fetch_add(base + 1, v.y, __ATOMIC_RELAXED, __HIP_MEMORY_SCOPE_AGENT);
    __hip_atomic_fetch_add(base + 2, v.z, __ATOMIC_RELAXED, __HIP_MEMORY_SCOPE_AGENT);
    __hip_atomic_fetch_add(base + 3, v.w, __ATOMIC_RELAXED, __HIP_MEMORY_SCOPE_AGENT);
    if ((gid & 31) == 0)
        __hip_atomic_fetch_add(deg + dst, 1.0f, __ATOMIC_RELAXED, __HIP_MEMORY_SCOPE_AGENT);
}

// ---------------- h_hat[b] = U[b]^T (64x2048) @ hb[b] (2048x128) ----------------
// one wave per (b, 16x16 output tile); K=2048 with 4 independent WMMA chains
__global__ __launch_bounds__(32) void k_hhat(const float* __restrict__ U,
                                             const float* __restrict__ h,
                                             float* __restrict__ hhat) {
    int bid = blockIdx.x;
    int b = bid >> 5, rem = bid & 31;
    int row0 = (rem >> 3) * 16;   // mode rows
    int col0 = (rem & 7) * 16;    // channel cols
    int lane = threadIdx.x, lm = lane & 15, kh = (lane >> 4) << 1;

    const float* up = U + (size_t)b * GNPG * GM;
    const float* hp = h + (size_t)b * GNPG * GC;

    v8f acc0 = {}, acc1 = {}, acc2 = {}, acc3 = {};
#define HH_STEP(KK, ACC)                                                         \
    {                                                                            \
        v2f a, bb;                                                               \
        a.x  = up[(size_t)((KK) + kh)     * GM + row0 + lm];                     \
        a.y  = up[(size_t)((KK) + kh + 1) * GM + row0 + lm];                     \
        bb.x = hp[(size_t)((KK) + kh)     * GC + col0 + lm];                     \
        bb.y = hp[(size_t)((KK) + kh + 1) * GC + col0 + lm];                     \
        ACC = wmma4(a, bb, ACC);                                                 \
    }
    for (int k = 0; k < GNPG; k += 16) {
        HH_STEP(k,      acc0);
        HH_STEP(k + 4,  acc1);
        HH_STEP(k + 8,  acc2);
        HH_STEP(k + 12, acc3);
    }
#undef HH_STEP
    v8f acc = (acc0 + acc1) + (acc2 + acc3);

    int rb = (lane >> 4) * 8;
    for (int r = 0; r < 8; ++r) {
        int m = row0 + rb + r;
        hhat[((size_t)b * GM + m) * GC + col0 + lm] = acc[r];
    }
}

// ---------------- out_hat[:,m,:] (64x128) = h_hat[:,m,:] @ W[m] (128x128) ----------------
__global__ __launch_bounds__(32) void k_outhat(const float* __restrict__ hhat,
                                               const float* __restrict__ W,
                                               float* __restrict__ outhat) {
    int bid = blockIdx.x;
    int m = bid >> 5, rem = bid & 31;
    int row0 = (rem >> 3) * 16;   // batch rows
    int col0 = (rem & 7) * 16;    // output channel cols
    int lane = threadIdx.x, lm = lane & 15, kh = (lane >> 4) << 1;

    const float* ap = hhat + ((size_t)(row0 + lm) * GM + m) * GC; // contiguous in k
    const float* wp = W + (size_t)m * GC * GC;

    v8f acc0 = {}, acc1 = {};
#define OH_STEP(KK, ACC)                                                         \
    {                                                                            \
        v2f a, bb;                                                               \
        a.x  = ap[(KK) + kh];                                                    \
        a.y  = ap[(KK) + kh + 1];                                                \
        bb.x = wp[(size_t)((KK) + kh)     * GC + col0 + lm];                     \
        bb.y = wp[(size_t)((KK) + kh + 1) * GC + col0 + lm];                     \
        ACC = wmma4(a, bb, ACC);                                                 \
    }
    for (int k = 0; k < GC; k += 8) {
        OH_STEP(k,     acc0);
        OH_STEP(k + 4, acc1);
    }
#undef OH_STEP
    v8f acc = acc0 + acc1;

    int rb = (lane >> 4) * 8;
    for (int r = 0; r < 8; ++r) {
        int row = row0 + rb + r;
        outhat[((size_t)row * GM + m) * GC + col0 + lm] = acc[r];
    }
}

// ---------------- fused: spec + local GEMMs -> +h +bias -> LayerNorm -> GELU ----------------
// one block = 16 node rows x full 128 channels; 8 waves, one 16x16 col-tile each
__global__ __launch_bounds__(256) void k_final(const float* __restrict__ h,
                                               const float* __restrict__ U,
                                               const float* __restrict__ outhat,
                                               const float* __restrict__ agg,
                                               const float* __restrict__ deg,
                                               const float* __restrict__ lin_w,
                                               const float* __restrict__ lin_b,
                                               const float* __restrict__ ln_g,
                                               const float* __restrict__ ln_beta,
                                               float* __restrict__ out) {
    __shared__ float xs[16][GC];
    __shared__ float ps[16][16], pq[16][16];
    __shared__ float mus[16], rsg[16];

    int node0 = blockIdx.x * 16;
    int b = node0 / GNPG;
    int n0 = node0 % GNPG;
    int wv = threadIdx.x >> 5;
    int lane = threadIdx.x & 31, lm = lane & 15, kh = (lane >> 4) << 1;
    int col0 = wv * 16;

    v8f acc = {};

    // spec contribution: U[b][n0:n0+16, :64] @ out_hat[b] (64x128), K=64
    const float* uap = U + ((size_t)b * GNPG + n0 + lm) * GM;   // contiguous in k
    const float* oh  = outhat + (size_t)b * GM * GC;
    for (int k = 0; k < GM; k += 4) {
        v2f a, bb;
        a.x  = uap[k + kh];
        a.y  = uap[k + kh + 1];
        bb.x = oh[(size_t)(k + kh)     * GC + col0 + lm];
        bb.y = oh[(size_t)(k + kh + 1) * GC + col0 + lm];
        acc = wmma4(a, bb, acc);
    }

    // local contribution: (agg/deg)[rows] @ lin_w^T, K=128 (B[k][n] = lin_w[n][k])
    const float* gap = agg + (size_t)(node0 + lm) * GC;
    float rdeg = 1.0f / fmaxf(deg[node0 + lm], 1.0f);
    for (int k = 0; k < GC; k += 4) {
        v2f a, bb;
        a.x  = gap[k + kh]     * rdeg;
        a.y  = gap[k + kh + 1] * rdeg;
        bb.x = lin_w[(size_t)(col0 + lm) * GC + k + kh];
        bb.y = lin_w[(size_t)(col0 + lm) * GC + k + kh + 1];
        acc = wmma4(a, bb, acc);
    }

    // x = h + spec + local(+lin_b) staged into LDS
    int rbase = (lane >> 4) * 8;
    float bias = lin_b[col0 + lm];
    for (int r = 0; r < 8; ++r) {
        int row = rbase + r;                           // local row 0..15
        float x = acc[r] + h[(size_t)(node0 + row) * GC + col0 + lm] + bias;
        xs[row][col0 + lm] = x;
    }
    __syncthreads();

    // LayerNorm reduction: thread t -> row t/16, 8-col segment t%16
    int r = threadIdx.x >> 4, seg = threadIdx.x & 15;
    float s = 0.0f, q = 0.0f;
    for (int i = 0; i < 8; ++i) {
        float v = xs[r][seg * 8 + i];
        s += v; q += v * v;
    }
    ps[r][seg] = s; pq[r][seg] = q;
    __syncthreads();
    if (threadIdx.x < 16) {
        float S = 0.0f, Q = 0.0f;
        for (int i = 0; i < 16; ++i) { S += ps[threadIdx.x][i]; Q += pq[threadIdx.x][i]; }
        float mu = S * (1.0f / GC);
        float var = Q * (1.0f / GC) - mu * mu;
        mus[threadIdx.x] = mu;
        rsg[threadIdx.x] = rsqrtf(var + 1e-5f);
    }
    __syncthreads();

    float mu = mus[r], rs = rsg[r];
    for (int i = 0; i < 8; ++i) {
        int c = seg * 8 + i;
        float xn = (xs[r][c] - mu) * rs * ln_g[c] + ln_beta[c];
        float g = 0.5f * xn * (1.0f + erff(xn * 0.70710678118654752f));
        out[(size_t)(node0 + r) * GC + c] = g;
    }
}

extern "C" void kernel_launch(void* const* d_in, const int* in_sizes, int n_in,
                              void* d_out, int out_size, void* d_ws, size_t ws_size,
                              hipStream_t stream) {
    const float* h       = (const float*)d_in[0];
    const float* U       = (const float*)d_in[1];
    const float* W       = (const float*)d_in[2];
    const float* lin_w   = (const float*)d_in[3];
    const float* lin_b   = (const float*)d_in[4];
    const float* ln_g    = (const float*)d_in[5];
    const float* ln_beta = (const float*)d_in[6];
    const int*   edge    = (const int*)d_in[7];
    float* out = (float*)d_out;
    int E = in_sizes[7] / 2;

    // workspace layout
    float* agg    = (float*)d_ws;                      // N*C
    float* deg    = agg    + (size_t)GN * GC;          // N
    float* hhat   = deg    + GN;                       // B*M*C
    float* outhat = hhat   + (size_t)GB * GM * GC;     // B*M*C

    long nz = (long)GN * GC + GN;
    k_zero<<<2048, 256, 0, stream>>>(agg, nz);

    long athreads = (long)E * 32;
    k_agg<<<(unsigned)((athreads + 255) / 256), 256, 0, stream>>>(h, edge, agg, deg, E);

    k_hhat<<<GB * 32, 32, 0, stream>>>(U, h, hhat);
    k_outhat<<<GM * 32, 32, 0, stream>>>(hhat, W, outhat);
    k_final<<<GN / 16, 256, 0, stream>>>(h, U, outhat, agg, deg,
                                         lin_w, lin_b, ln_g, ln_beta, out);
}